// GlobalPooling_67542655697484
// MI455X (gfx1250) — compile-verified
//
#include <hip/hip_runtime.h>
#include <math.h>

#define HDIM 256
#define BSEG 4096
#define ROWS 256   // rows per pooling block

typedef __attribute__((ext_vector_type(2))) float v2f;
typedef __attribute__((ext_vector_type(8))) float v8f;

// ---- ordered-uint key for float atomic max (monotone bijection) ----
__device__ __forceinline__ unsigned fkey(float f) {
    unsigned u = __float_as_uint(f);
    return (u & 0x80000000u) ? (~u) : (u | 0x80000000u);
}
__device__ __forceinline__ float funkey(unsigned k) {
    unsigned u = (k & 0x80000000u) ? (k & 0x7FFFFFFFu) : (~k);
    return __uint_as_float(u);
}

// ---------------- init workspace ----------------
__global__ void gp_init_kernel(float* __restrict__ sum, unsigned* __restrict__ maxkey,
                               float* __restrict__ counts, int bh, int b) {
    int i = blockIdx.x * blockDim.x + threadIdx.x;
    if (i < bh) { sum[i] = 0.0f; maxkey[i] = 0u; }   // 0 < fkey(-inf) = 0x007FFFFF
    if (i < b)  counts[i] = 0.0f;
}

// ---------------- segment pooling (streaming, sorted batch) ----------------
__global__ void gp_pool_kernel(const float* __restrict__ feat, const int* __restrict__ batch,
                               float* __restrict__ sum, unsigned* __restrict__ maxkey,
                               float* __restrict__ counts, int n) {
    __shared__ int sseg[ROWS];
    const int c = threadIdx.x;                 // column 0..255
    const long long r0 = (long long)blockIdx.x * ROWS;
    if (r0 >= n) return;
    const int nr = min(ROWS, n - (int)r0);
    if (c < nr) sseg[c] = batch[r0 + c];
    __syncthreads();

    float rs = 0.0f;
    float rm = -__builtin_inff();
    int   cur = sseg[0];
    int   runLen = 0;
    for (int i = 0; i < nr; ++i) {
        const int s = sseg[i];
        if (s != cur) {
            atomicAdd(&sum[(long long)cur * HDIM + c], rs);
            atomicMax(&maxkey[(long long)cur * HDIM + c], fkey(rm));
            if (c == 0) atomicAdd(&counts[cur], (float)runLen);
            rs = 0.0f; rm = -__builtin_inff(); runLen = 0; cur = s;
        }
        const float v = __builtin_nontemporal_load(&feat[(r0 + i) * HDIM + c]);
        rs += v;
        rm = fmaxf(rm, v);
        ++runLen;
    }
    atomicAdd(&sum[(long long)cur * HDIM + c], rs);
    atomicMax(&maxkey[(long long)cur * HDIM + c], fkey(rm));
    if (c == 0) atomicAdd(&counts[cur], (float)runLen);
}

// ---------------- GEMM1: silu(combined @ W1 + b1) -> h [4096,256] ----------------
// one wave (32 threads) per 16x16 output tile; f32 WMMA 16x16x4.
// K-loop split by region (mean | sum | max) so the hot loops have no selects.
__global__ void gp_gemm1_kernel(const float* __restrict__ sum, const unsigned* __restrict__ maxkey,
                                const float* __restrict__ counts,
                                const float* __restrict__ W1, const float* __restrict__ b1,
                                float* __restrict__ h) {
    const int lane = threadIdx.x;            // 0..31
    const int tile = blockIdx.x;             // 4096 tiles
    const int tm = tile >> 4;                // row tile (B/16 = 256)
    const int tn = tile & 15;                // col tile (H/16 = 16)
    const int l15 = lane & 15;
    const int kg  = lane >> 4;               // K-pair group (0/1)
    const int arow = tm * 16 + l15;          // A: M = lane&15
    const int bcol = tn * 16 + l15;          // B: N = lane&15
    const float cnt = counts[arow];
    const float inv = 1.0f / fmaxf(cnt, 1.0f);
    const bool nonempty = (cnt > 0.0f);

    const float*    srow = sum    + (long long)arow * HDIM;
    const unsigned* mrow = maxkey + (long long)arow * HDIM;

    v8f acc = {};

    // region 0: mean = sum * inv, W1 rows [0,256)
    for (int k0 = 0; k0 < HDIM; k0 += 4) {
        const int kA = k0 + kg * 2;
        v2f a, b;
        a.x = srow[kA]     * inv;
        a.y = srow[kA + 1] * inv;
        b.x = W1[(long long)kA * HDIM + bcol];
        b.y = W1[(long long)(kA + 1) * HDIM + bcol];
        acc = __builtin_amdgcn_wmma_f32_16x16x4_f32(false, a, false, b,
                                                    (short)0, acc, false, false);
    }
    // region 1: sum, W1 rows [256,512)
    for (int k0 = 0; k0 < HDIM; k0 += 4) {
        const int kA = k0 + kg * 2;
        v2f a, b;
        a.x = srow[kA];
        a.y = srow[kA + 1];
        b.x = W1[(long long)(HDIM + kA) * HDIM + bcol];
        b.y = W1[(long long)(HDIM + kA + 1) * HDIM + bcol];
        acc = __builtin_amdgcn_wmma_f32_16x16x4_f32(false, a, false, b,
                                                    (short)0, acc, false, false);
    }
    // region 2: max (decoded; empty segment -> 0), W1 rows [512,768)
    for (int k0 = 0; k0 < HDIM; k0 += 4) {
        const int kA = k0 + kg * 2;
        v2f a, b;
        a.x = nonempty ? funkey(mrow[kA])     : 0.0f;   // per-lane select, no branch
        a.y = nonempty ? funkey(mrow[kA + 1]) : 0.0f;
        b.x = W1[(long long)(2 * HDIM + kA) * HDIM + bcol];
        b.y = W1[(long long)(2 * HDIM + kA + 1) * HDIM + bcol];
        acc = __builtin_amdgcn_wmma_f32_16x16x4_f32(false, a, false, b,
                                                    (short)0, acc, false, false);
    }

    // D layout: VGPR r -> M = r + 8*(lane>=16), N = lane&15
    const int ncol  = tn * 16 + l15;
    const int rbase = tm * 16 + (kg << 3);
    const float bias = b1[ncol];
    #pragma unroll
    for (int r = 0; r < 8; ++r) {
        float v = acc[r] + bias;
        float sv = v / (1.0f + expf(-v));    // silu
        h[(long long)(rbase + r) * HDIM + ncol] = sv;
    }
}

// ---------------- GEMM2: h @ W2 + b2 -> out [4096,256] ----------------
__global__ void gp_gemm2_kernel(const float* __restrict__ h,
                                const float* __restrict__ W2, const float* __restrict__ b2,
                                float* __restrict__ out) {
    const int lane = threadIdx.x;
    const int tile = blockIdx.x;
    const int tm = tile >> 4;
    const int tn = tile & 15;
    const int l15 = lane & 15;
    const int kg  = lane >> 4;
    const int arow = tm * 16 + l15;
    const int bcol = tn * 16 + l15;

    const float* hrow = h + (long long)arow * HDIM;

    v8f acc = {};
    for (int k0 = 0; k0 < HDIM; k0 += 4) {
        const int kA = k0 + kg * 2;
        v2f a, b;
        a.x = hrow[kA];
        a.y = hrow[kA + 1];
        b.x = W2[(long long)kA * HDIM + bcol];
        b.y = W2[(long long)(kA + 1) * HDIM + bcol];
        acc = __builtin_amdgcn_wmma_f32_16x16x4_f32(false, a, false, b,
                                                    (short)0, acc, false, false);
    }

    const int ncol  = tn * 16 + l15;
    const int rbase = tm * 16 + (kg << 3);
    const float bias = b2[ncol];
    #pragma unroll
    for (int r = 0; r < 8; ++r) {
        out[(long long)(rbase + r) * HDIM + ncol] = acc[r] + bias;
    }
}

extern "C" void kernel_launch(void* const* d_in, const int* in_sizes, int n_in,
                              void* d_out, int out_size, void* d_ws, size_t ws_size,
                              hipStream_t stream) {
    const float* feat = (const float*)d_in[0];
    const int*   batch = (const int*)d_in[1];
    const float* W1 = (const float*)d_in[2];
    const float* b1 = (const float*)d_in[3];
    const float* W2 = (const float*)d_in[4];
    const float* b2 = (const float*)d_in[5];
    float* out = (float*)d_out;

    const int n = in_sizes[0] / HDIM;        // N atoms
    const int bh = BSEG * HDIM;              // 1,048,576

    char* ws = (char*)d_ws;
    float*    sum    = (float*)ws;                                   // 4 MB
    unsigned* maxkey = (unsigned*)(ws + (size_t)bh * 4);             // 4 MB
    float*    hbuf   = (float*)(ws + (size_t)bh * 8);                // 4 MB
    float*    counts = (float*)(ws + (size_t)bh * 12);               // 16 KB

    gp_init_kernel<<<(bh + 255) / 256, 256, 0, stream>>>(sum, maxkey, counts, bh, BSEG);
    gp_pool_kernel<<<(n + ROWS - 1) / ROWS, HDIM, 0, stream>>>(feat, batch, sum, maxkey, counts, n);
    gp_gemm1_kernel<<<(BSEG / 16) * (HDIM / 16), 32, 0, stream>>>(sum, maxkey, counts, W1, b1, hbuf);
    gp_gemm2_kernel<<<(BSEG / 16) * (HDIM / 16), 32, 0, stream>>>(hbuf, W2, b2, out);
}